// TemporalPointProcessOnGraph_24077586661973
// MI455X (gfx1250) — compile-verified
//
#include <hip/hip_runtime.h>
#include <hip/hip_bf16.h>

#define TAU_MAX 0.1f
#define SEQ_LEN 2048           // reference S (B*S = in_sizes[0])
#define PP 4                   // n_basis_time
#define QQ 4                   // n_basis_loc
#define RMAX 128               // >= eval_res (100)

typedef float v2f __attribute__((ext_vector_type(2)));
typedef float v4f __attribute__((ext_vector_type(4)));
typedef float v8f __attribute__((ext_vector_type(8)));

// ---------------------------------------------------------------------------
// Kernel 1: WBf[pair][p] = sum_q W[p,q] * Bfilt[q][pair]   (pair-major, v4)
// One V_WMMA_F32_16X16X4_F32 per wave per 16-column tile:
//   A (16x4)  = W padded to 16 rows (rows 4..15 zero)
//   B (4x16)  = Bfilt[:, tile]
//   D (16x16) rows 0..3 = WBf[:, tile] -> stored transposed as float4/column
// ---------------------------------------------------------------------------
__global__ void __launch_bounds__(256)
wmix_precompute_kernel(const float* __restrict__ W,
                       const float* __restrict__ Bfilt,
                       float* __restrict__ WBf,   // [NN][4]
                       int NN /* = N*N */, int ntiles)
{
    const int lane = threadIdx.x & 31;
    const int row  = lane & 15;          // M index (A) / N index (B,D)
    const int kb   = (lane >> 4) << 1;   // K base: 0 for lanes 0-15, 2 for 16-31

    const int wave = (blockIdx.x * blockDim.x + threadIdx.x) >> 5;
    if (wave >= ntiles) return;          // wave-uniform: EXEC stays all-ones

    // A matrix: W (4x4) in rows 0..3, zeros elsewhere
    v2f a;
    a.x = (row < PP) ? W[row * QQ + kb]     : 0.0f;
    a.y = (row < PP) ? W[row * QQ + kb + 1] : 0.0f;

    const int base = wave * 16;
    int col  = base + row;
    int colc = (col < NN) ? col : (NN - 1);   // branchless clamp, keep EXEC full

    // B matrix: rows K=kb, kb+1 at column `col`
    v2f bm;
    bm.x = Bfilt[(size_t)kb       * NN + colc];
    bm.y = Bfilt[(size_t)(kb + 1) * NN + colc];

    v8f c = {};
    v8f d = __builtin_amdgcn_wmma_f32_16x16x4_f32(
        /*neg_a=*/false, a, /*neg_b=*/false, bm,
        /*c_mod=*/(short)0, c, /*reuse_a=*/false, /*reuse_b=*/false);

    // D rows 0..3, lanes 0..15 hold WBf[p][col]; store pair-major as one b128
    if (lane < 16 && col < NN) {
        v4f o; o[0] = d[0]; o[1] = d[1]; o[2] = d[2]; o[3] = d[3];
        *(v4f*)(WBf + (size_t)col * 4) = o;
    }
}

// ---------------------------------------------------------------------------
// Kernel 1b: transpose ybasis [P][BS] -> ybt [BS][4] (one b128 store / event)
// ---------------------------------------------------------------------------
__global__ void __launch_bounds__(256)
ybasis_transpose_kernel(const float* __restrict__ ybasis,
                        float* __restrict__ ybt, int BS)
{
    int e = blockIdx.x * blockDim.x + threadIdx.x;
    if (e >= BS) return;
    v4f o;
    #pragma unroll
    for (int p = 0; p < PP; ++p) o[p] = ybasis[(size_t)p * BS + e];
    *(v4f*)(ybt + (size_t)e * 4) = o;
}

// ---------------------------------------------------------------------------
// Kernel 2: intensity. One wave32 per event i; lanes stride backward over the
// tau-window (ts sorted => tau monotone => per-lane early break is exact).
// USE_WBF: one b128 gather (WBf) + one b128 load (ybt) + two ds_load_b128
// per pair. Else: fused W-mix with scalar gathers of Bfilt (input layout).
// ---------------------------------------------------------------------------
template <bool USE_WBF>
__global__ void __launch_bounds__(256)
tpp_intensity_kernel(const float* __restrict__ ts,
                     const int*   __restrict__ ls,
                     const float* __restrict__ mu,
                     const float* __restrict__ xgrids,
                     const float* __restrict__ ybasis,  // [P][BS] (fused path)
                     const float* __restrict__ ybt,     // [BS][4] (WBF path)
                     const float* __restrict__ W,
                     const float* __restrict__ G,       // WBf[NN][4] or Bfilt[Q][NN]
                     float* __restrict__ out,
                     int S, int N, int R, int NN, int BS)
{
    // time-basis grids staged in LDS, r-major [r][4] for 16B vector loads
    __shared__ float xs[RMAX * PP];

    for (int idx = threadIdx.x; idx < PP * R; idx += blockDim.x) {
        const int p = idx & 3, r = idx >> 2;
        xs[idx] = xgrids[p * R + r];
    }
    __syncthreads();

    const int wv   = threadIdx.x >> 5;
    const int lane = threadIdx.x & 31;
    const int e    = blockIdx.x * 8 + wv;          // global event index
    if (e >= BS) return;

    const int b   = e / S;
    const int i   = e - b * S;
    const int rb  = b * S;                          // row base
    const float ti = ts[e];
    const int   li = ls[e];

    // fused path: W in registers (broadcast loads, once)
    float Wr[PP * QQ];
    if (!USE_WBF) {
        #pragma unroll
        for (int k = 0; k < PP * QQ; ++k) Wr[k] = W[k];
    }

    float acc = 0.0f;
    if (ti > 0.0f) {
        const float inv_h = (float)(R - 1) / TAU_MAX;
        for (int j = i - 1 - lane; j >= 0; j -= 32) {
            const int  ej  = rb + j;
            const float tj  = ts[ej];
            const float tau = ti - tj;
            if (tau > TAU_MAX) break;               // earlier j only larger tau
            if (tau > 0.0f) {                       // skip exact ties
                float pos = tau * inv_h;            // in (0, R-1]
                int i0 = (int)pos;
                if (i0 > R - 2) i0 = R - 2;
                const float frac = pos - (float)i0;

                const int pair = ls[ej] * N + li;

                if (USE_WBF) {
                    const v4f x0  = *(const v4f*)(xs + i0 * 4);       // ds b128
                    const v4f x1  = *(const v4f*)(xs + i0 * 4 + 4);   // ds b128
                    const v4f yb  = *(const v4f*)(ybt + (size_t)ej * 4);
                    const v4f wb  = *(const v4f*)(G + (size_t)pair * 4);
                    #pragma unroll
                    for (int p = 0; p < PP; ++p) {
                        const float xv = x0[p] + frac * (x1[p] - x0[p]);
                        acc += xv * yb[p] * wb[p];
                    }
                } else {
                    float ap[PP];
                    #pragma unroll
                    for (int p = 0; p < PP; ++p) {
                        const float a0 = xs[i0 * 4 + p];
                        const float a1 = xs[i0 * 4 + 4 + p];
                        const float xv = a0 + frac * (a1 - a0);
                        ap[p] = xv * ybasis[(size_t)p * BS + ej];
                    }
                    float K = 0.0f;
                    #pragma unroll
                    for (int q = 0; q < QQ; ++q) {
                        float cq = 0.0f;
                        #pragma unroll
                        for (int p = 0; p < PP; ++p)
                            cq += Wr[p * QQ + q] * ap[p];
                        K += cq * G[(size_t)q * NN + pair];
                    }
                    acc += K;
                }
            }
        }
    }

    // wave32 reduction
    #pragma unroll
    for (int off = 16; off > 0; off >>= 1)
        acc += __shfl_down(acc, off, 32);

    if (lane == 0)
        out[e] = (ti > 0.0f) ? (mu[li] + acc) : 0.0f;
}

// ---------------------------------------------------------------------------
extern "C" void kernel_launch(void* const* d_in, const int* in_sizes, int n_in,
                              void* d_out, int out_size, void* d_ws, size_t ws_size,
                              hipStream_t stream)
{
    const float* ts     = (const float*)d_in[0];
    const int*   ls     = (const int*)  d_in[1];
    const float* mu     = (const float*)d_in[2];
    const float* xgrids = (const float*)d_in[3];
    const float* ybasis = (const float*)d_in[4];
    const float* W      = (const float*)d_in[5];
    const float* Bfilt  = (const float*)d_in[6];
    float*       out    = (float*)d_out;

    const int BS = in_sizes[0];            // B*S
    const int N  = in_sizes[2];            // n_node
    const int P  = in_sizes[4] / BS;       // n_basis_time (== 4)
    const int R  = in_sizes[3] / P;        // eval_res (== 100)
    const int S  = SEQ_LEN;                // reference seq len
    const int NN = N * N;

    const int mainBlocks = (BS + 7) / 8;   // 8 waves of 32 = 8 events per block

    const size_t wbfBytes = (size_t)PP * (size_t)NN * sizeof(float);  // 16B-mult
    const size_t ybtBytes = (size_t)PP * (size_t)BS * sizeof(float);
    const size_t needWs   = wbfBytes + ybtBytes;

    if (ws_size >= needWs) {
        float* WBf = (float*)d_ws;
        float* ybt = (float*)((char*)d_ws + wbfBytes);

        // Phase 1a: hoist the W-mix out of the pairwise loop via WMMA
        const int ntiles = (NN + 15) / 16;
        const int blocks = (ntiles + 7) / 8;       // 8 waves (tiles) per block
        wmix_precompute_kernel<<<blocks, 256, 0, stream>>>(W, Bfilt, WBf, NN, ntiles);
        // Phase 1b: transpose ybasis to event-major float4
        ybasis_transpose_kernel<<<(BS + 255) / 256, 256, 0, stream>>>(ybasis, ybt, BS);
        // Phase 2: windowed pairwise intensity
        tpp_intensity_kernel<true><<<mainBlocks, 256, 0, stream>>>(
            ts, ls, mu, xgrids, ybasis, ybt, W, WBf, out, S, N, R, NN, BS);
    } else {
        tpp_intensity_kernel<false><<<mainBlocks, 256, 0, stream>>>(
            ts, ls, mu, xgrids, ybasis, nullptr, W, Bfilt, out, S, N, R, NN, BS);
    }
}